// RelativeAttention_5274219839857
// MI455X (gfx1250) — compile-verified
//
#include <hip/hip_runtime.h>

// ---------------------------------------------------------------------------
// Relative attention (music-transformer style) for MI455X / gfx1250.
//  B=8, N=2048, D=512, fp32.  d_out = [out: B*N*D][p_attn: B*N*N] (fp32).
//
// Pipeline:
//  K1 scores_kernel : S = Q@K^T + skew(Q@Wk_band^T)   -> p region (raw scores)
//  K2 softmax_kernel: row softmax in place             -> p region (p_attn)
//  K3 out_kernel    : out = P@V + shift(P)@Wv_band     -> out region
//
// All matmuls via V_WMMA_F32_16X16X4_F32 (fp32 accumulate, matches reference
// precision).  Toeplitz structure of the relative term is exploited so the
// rel GEMMs cost the same as the dense ones (no [B,N,2N-1] blowup).
// K1 uses 4 independent WMMA accumulator chains per wave (even/odd k-step
// split for both the QK^T chain and the Toeplitz-T chain) to hide WMMA
// latency; K3 gets its ILP from 8 per-d-tile chains.
// ---------------------------------------------------------------------------

typedef float v2f __attribute__((ext_vector_type(2)));
typedef float v4f __attribute__((ext_vector_type(4)));
typedef float v8f __attribute__((ext_vector_type(8)));

#define B_ 8
#define N_ 2048
#define D_ 512
#define RMAX_ (2 * N_ - 2) /* max valid row of w_k / w_v (2N-1 rows) */

__device__ __forceinline__ v8f wmma_f32_k4(v2f a, v2f b, v8f c) {
  // D = A(16x4 f32) * B(4x16 f32) + C(16x16 f32)
  return __builtin_amdgcn_wmma_f32_16x16x4_f32(false, a, false, b, (short)0, c,
                                               false, false);
}

__device__ __forceinline__ v8f vzero8() {
  v8f z = {0.f, 0.f, 0.f, 0.f, 0.f, 0.f, 0.f, 0.f};
  return z;
}

// ---------------------------------------------------------------------------
// Kernel 1: raw scores.  Grid (N/64, B), block 128 (4 waves).
// Wave w owns a 16-row i-tile (i0 = 64*blockIdx.x + 16*w) and sweeps all 128
// j-tiles.  Per j-tile: base = Q@K^T fused with the next sliding Toeplitz
// tile T[:,16(m+1)..] = Q@Wk_band^T (shared A operand).  Each GEMM uses an
// even/odd k-step accumulator pair -> 4 independent WMMA chains in flight.
// rel is assembled from the per-wave 16x32 T window in LDS via the
// anti-diagonal gather.
// ---------------------------------------------------------------------------
__global__ __launch_bounds__(128) void scores_kernel(
    const float* __restrict__ Q, const float* __restrict__ K,
    const float* __restrict__ Wk, float* __restrict__ S) {
  const int b = blockIdx.y;
  const int gi0 = blockIdx.x * 64;
  const int tid = threadIdx.x;
  const int wv = tid >> 5;
  const int ln = tid & 31;
  const int half = ln >> 4; // 0: lanes 0-15, 1: lanes 16-31
  const int lr = ln & 15;

  __shared__ float Qs[64][D_ + 4]; // pad 4 -> conflict-free b64 reads, 16B ok
  __shared__ float Tc[4][16][34];  // per-wave 16x32 circular T window

  // cooperative load of 64 Q rows (b128 both sides; stride 516 keeps 16B)
  const float* Qg = Q + ((size_t)b * N_ + gi0) * D_;
  for (int idx = tid; idx < 64 * (D_ / 4); idx += 128) {
    const int row = idx / (D_ / 4);
    const int cp = (idx % (D_ / 4)) * 4;
    *(v4f*)&Qs[row][cp] = *(const v4f*)(Qg + (size_t)row * D_ + cp);
  }
  __syncthreads();

  const int i0 = gi0 + wv * 16;
  const int cbase = N_ - 16 - i0; // w_k band base: r = cbase + c, c=j+15-ii
  const float* Kg = K + (size_t)b * N_ * D_;
  const int kcol = 2 * half; // this lane's K-dim offset within a 4-slice

  // ---- prologue: T c-tile 0 ----
  {
    v8f t0 = vzero8(), t1 = vzero8();
    int r = cbase + lr;
    if (r > RMAX_) r = RMAX_;
    const float* wrow = Wk + (size_t)r * D_;
    const float* qrow = &Qs[wv * 16 + lr][0];
#pragma unroll 2
    for (int k0 = 0; k0 < D_; k0 += 8) {
      const v2f a0 = *(const v2f*)(qrow + k0 + kcol);
      const v2f a1 = *(const v2f*)(qrow + k0 + 4 + kcol);
      const v2f b0 = *(const v2f*)(wrow + k0 + kcol);
      const v2f b1 = *(const v2f*)(wrow + k0 + 4 + kcol);
      t0 = wmma_f32_k4(a0, b0, t0);
      t1 = wmma_f32_k4(a1, b1, t1);
    }
    t0 = t0 + t1;
#pragma unroll
    for (int v = 0; v < 8; ++v) Tc[wv][v + 8 * half][lr] = t0[v];
  }

  float* Srow = S + ((size_t)b * N_ + i0) * N_;

  for (int m = 0; m < N_ / 16; ++m) {
    const int j0 = m * 16;
    const int c0 = j0 + 16; // new T tile base
    v8f acc0 = vzero8(), acc1 = vzero8(); // QK^T even/odd chains
    v8f tn0 = vzero8(), tn1 = vzero8();   // T even/odd chains

    int r = cbase + c0 + lr;
    if (r > RMAX_) r = RMAX_; // pad column: never read back
    const float* wrow = Wk + (size_t)r * D_;
    const float* krow = Kg + (size_t)(j0 + lr) * D_;
    const float* qrow = &Qs[wv * 16 + lr][0];

#pragma unroll 2
    for (int k0 = 0; k0 < D_; k0 += 8) {
      const v2f a0 = *(const v2f*)(qrow + k0 + kcol);
      const v2f a1 = *(const v2f*)(qrow + k0 + 4 + kcol);
      const v2f bk0 = *(const v2f*)(krow + k0 + kcol);
      const v2f bk1 = *(const v2f*)(krow + k0 + 4 + kcol);
      const v2f bw0 = *(const v2f*)(wrow + k0 + kcol);
      const v2f bw1 = *(const v2f*)(wrow + k0 + 4 + kcol);
      acc0 = wmma_f32_k4(a0, bk0, acc0);
      tn0 = wmma_f32_k4(a0, bw0, tn0);
      acc1 = wmma_f32_k4(a1, bk1, acc1);
      tn1 = wmma_f32_k4(a1, bw1, tn1);
    }
    acc0 = acc0 + acc1;
    tn0 = tn0 + tn1;

    // store new T tile into its circular half
#pragma unroll
    for (int v = 0; v < 8; ++v)
      Tc[wv][v + 8 * half][(c0 + lr) & 31] = tn0[v];

    // assemble rel via anti-diagonal gather and write the score tile
    const int jj = lr;
#pragma unroll
    for (int v = 0; v < 8; ++v) {
      const int ii = v + 8 * half;
      const float rel = Tc[wv][ii][(j0 + jj + 15 - ii) & 31];
      Srow[(size_t)ii * N_ + j0 + jj] = acc0[v] + rel;
    }
  }
}

// ---------------------------------------------------------------------------
// Kernel 2: row softmax in place.  Grid B*N, block 256 (8 elems/thread).
// ---------------------------------------------------------------------------
__global__ __launch_bounds__(256) void softmax_kernel(float* __restrict__ P) {
  float* p = P + (size_t)blockIdx.x * N_;
  const int t = threadIdx.x;
  __shared__ float red[256];

  float v[8];
  float mx = -3.402823466e38f;
#pragma unroll
  for (int q = 0; q < 8; ++q) {
    v[q] = p[t + 256 * q];
    mx = fmaxf(mx, v[q]);
  }
  red[t] = mx;
  __syncthreads();
  for (int s = 128; s > 0; s >>= 1) {
    if (t < s) red[t] = fmaxf(red[t], red[t + s]);
    __syncthreads();
  }
  mx = red[0];
  __syncthreads();

  float sum = 0.f;
#pragma unroll
  for (int q = 0; q < 8; ++q) {
    v[q] = __expf(v[q] - mx);
    sum += v[q];
  }
  red[t] = sum;
  __syncthreads();
  for (int s = 128; s > 0; s >>= 1) {
    if (t < s) red[t] += red[t + s];
    __syncthreads();
  }
  const float inv = 1.f / red[0];
#pragma unroll
  for (int q = 0; q < 8; ++q) p[t + 256 * q] = v[q] * inv;
}

// ---------------------------------------------------------------------------
// Kernel 3: out = P@V + shift(P)@Wv_band.  Grid (N/16, B), block 128.
// Workgroup owns one 16-row i-tile; wave w owns d-columns [128w, 128w+128)
// (8 d-tiles, one fused accumulator each -> 8 independent WMMA chains).
// Reduction over j (PV) and over the Toeplitz index c (rel; A-operand
// row-shifted reads of P, length N+15 padded to N+16).  w_v row clamped at
// 2N-2; the pad column has A'==0.
// ---------------------------------------------------------------------------
__global__ __launch_bounds__(128) void out_kernel(
    const float* __restrict__ P, const float* __restrict__ V,
    const float* __restrict__ Wv, float* __restrict__ O) {
  const int b = blockIdx.y;
  const int i0 = blockIdx.x * 16;
  const int tid = threadIdx.x;
  const int wv = tid >> 5;
  const int ln = tid & 31;
  const int half = ln >> 4;
  const int lr = ln & 15;
  const int kcol = 2 * half;

  const int cbase = N_ - 16 - i0;
  const float* Prow = P + ((size_t)b * N_ + i0) * N_; // 16 x N
  const float* Vg = V + (size_t)b * N_ * D_;
  const int dbase = wv * 128;

  v8f acc[8];
#pragma unroll
  for (int dt = 0; dt < 8; ++dt) acc[dt] = vzero8();

  for (int m = 0; m < N_ / 16; ++m) {
#pragma unroll
    for (int s = 0; s < 4; ++s) {
      const int k0 = m * 16 + s * 4;
      const float* prow = Prow + (size_t)lr * N_;
      // PV A-slice: p[i0+lr, k0+kcol .. +1]
      const v2f aPV = *(const v2f*)(prow + k0 + kcol);
      // rel A-slice: p[i0+lr, c+lr-15] with bounds -> 0
      const int jr = k0 + kcol + lr - 15;
      v2f aR;
      aR.x = (jr >= 0 && jr < N_) ? prow[jr] : 0.f;
      aR.y = (jr + 1 >= 0 && jr + 1 < N_) ? prow[jr + 1] : 0.f;
      // w_v band rows (clamped; pad has aR==0)
      const int r0 = cbase + k0 + kcol;
      const int r0c = r0 > RMAX_ ? RMAX_ : r0;
      const int r1c = (r0 + 1) > RMAX_ ? RMAX_ : (r0 + 1);
      const float* v0 = Vg + (size_t)(k0 + kcol) * D_;
      const float* v1 = v0 + D_;
      const float* w0 = Wv + (size_t)r0c * D_;
      const float* w1 = Wv + (size_t)r1c * D_;
#pragma unroll
      for (int dt = 0; dt < 8; ++dt) {
        const int d0 = dbase + dt * 16 + lr;
        v2f bV;
        bV.x = v0[d0];
        bV.y = v1[d0];
        v2f bW;
        bW.x = w0[d0];
        bW.y = w1[d0];
        acc[dt] = wmma_f32_k4(aPV, bV, acc[dt]);
        acc[dt] = wmma_f32_k4(aR, bW, acc[dt]);
      }
    }
  }

  // epilogue: rel c-tile m=128 (c in [2048,2064), only c<=2062 has aR!=0)
#pragma unroll
  for (int s = 0; s < 4; ++s) {
    const int k0 = N_ + s * 4;
    const float* prow = Prow + (size_t)lr * N_;
    const int jr = k0 + kcol + lr - 15; // >= 2048-15 >= 0
    v2f aR;
    aR.x = (jr < N_) ? prow[jr] : 0.f;
    aR.y = (jr + 1 < N_) ? prow[jr + 1] : 0.f;
    const int r0 = cbase + k0 + kcol;
    const int r0c = r0 > RMAX_ ? RMAX_ : r0;
    const int r1c = (r0 + 1) > RMAX_ ? RMAX_ : (r0 + 1);
    const float* w0 = Wv + (size_t)r0c * D_;
    const float* w1 = Wv + (size_t)r1c * D_;
#pragma unroll
    for (int dt = 0; dt < 8; ++dt) {
      const int d0 = dbase + dt * 16 + lr;
      v2f bW;
      bW.x = w0[d0];
      bW.y = w1[d0];
      acc[dt] = wmma_f32_k4(aR, bW, acc[dt]);
    }
  }

  float* Orow = O + ((size_t)b * N_ + i0) * D_;
#pragma unroll
  for (int dt = 0; dt < 8; ++dt) {
#pragma unroll
    for (int v = 0; v < 8; ++v) {
      const int ii = v + 8 * half;
      Orow[(size_t)ii * D_ + dbase + dt * 16 + lr] = acc[dt][v];
    }
  }
}

// ---------------------------------------------------------------------------
extern "C" void kernel_launch(void* const* d_in, const int* in_sizes, int n_in,
                              void* d_out, int out_size, void* d_ws,
                              size_t ws_size, hipStream_t stream) {
  (void)in_sizes;
  (void)n_in;
  (void)out_size;
  (void)d_ws;
  (void)ws_size;

  const float* Q = (const float*)d_in[0];
  const float* K = (const float*)d_in[1];
  const float* V = (const float*)d_in[2];
  const float* Wk = (const float*)d_in[3];
  const float* Wv = (const float*)d_in[4];

  float* Out = (float*)d_out;             // [B, N, D]
  float* Pp = Out + (size_t)B_ * N_ * D_; // [B, N, N] scores -> p_attn

  scores_kernel<<<dim3(N_ / 64, B_), 128, 0, stream>>>(Q, K, Wk, Pp);
  softmax_kernel<<<dim3(B_ * N_), 256, 0, stream>>>(Pp);
  out_kernel<<<dim3(N_ / 16, B_), 128, 0, stream>>>(Pp, V, Wv, Out);
}